// PolicyNetRegression_36756330119246
// MI455X (gfx1250) — compile-verified
//
#include <hip/hip_runtime.h>
#include <math.h>

// ---------------------------------------------------------------------------
// PolicyNetRegression for MI455X (gfx1250, wave32, WMMA f32 16x16x4)
// ---------------------------------------------------------------------------

typedef __attribute__((ext_vector_type(2))) float v2f;
typedef __attribute__((ext_vector_type(8))) float v8f;

constexpr int kB  = 8;
constexpr int kS  = 20;
constexpr int kL  = 100;
constexpr int kLL = 200;
constexpr int kU  = 100;
constexpr int kE  = 64;
constexpr int kH  = 256;
constexpr int kV  = 100000;
constexpr int kK  = 100;
constexpr int k4H = 4 * kH;    // 1024
constexpr int kN5 = 5 * kE;    // 320
constexpr int kNU = kU * kE;   // 6400

__device__ __forceinline__ float sigmoidf(float x) { return 1.0f / (1.0f + expf(-x)); }

// D = A(16x4 f32) * B(4x16 f32) + C(16x16 f32), exact f32 per CDNA5 ISA.
__device__ __forceinline__ v8f wmma4(v8f acc, v2f a, v2f b) {
    return __builtin_amdgcn_wmma_f32_16x16x4_f32(false, a, false, b, (short)0, acc,
                                                 false, false);
}

// Monotone float -> uint key (larger float => larger key).
__device__ __forceinline__ unsigned fkey(float f) {
    unsigned u = __float_as_uint(f);
    return (u & 0x80000000u) ? ~u : (u | 0x80000000u);
}

// ---------------------------------------------------------------------------
// 1. x[b,s,:] = graph_emb[inputs[b,s] * mask[b,s]]
// ---------------------------------------------------------------------------
__global__ void k_gather(const int* __restrict__ inputs, const int* __restrict__ mask,
                         const float* __restrict__ graph_emb, float* __restrict__ x) {
    int t = blockIdx.x * blockDim.x + threadIdx.x;
    if (t >= kB * kS * kE) return;
    int e = t % kE, bs = t / kE;
    int id = inputs[bs] * mask[bs];
    x[t] = graph_emb[id * kE + e];
}

// ---------------------------------------------------------------------------
// 2. Two-layer LSTM, single block (sequential over S; L2-resident weights).
//    h-state in LDS (shared across threads); c-state in registers (private).
// ---------------------------------------------------------------------------
__device__ __forceinline__ float lstm_update(const float* gb, int u, float& c) {
    float gi = gb[u], gf = gb[kH + u], gg = gb[2 * kH + u], go = gb[3 * kH + u];
    c = sigmoidf(gf) * c + sigmoidf(gi) * tanhf(gg);
    return sigmoidf(go) * tanhf(c);
}

__global__ void __launch_bounds__(1024)
k_lstm(const float* __restrict__ x,
       const float* __restrict__ Wih0, const float* __restrict__ Whh0,
       const float* __restrict__ bih0, const float* __restrict__ bhh0,
       const float* __restrict__ Wih1, const float* __restrict__ Whh1,
       const float* __restrict__ bih1, const float* __restrict__ bhh1,
       float* __restrict__ outseq, float* __restrict__ h2) {
    __shared__ float sX[kB * kE];    //  2 KB
    __shared__ float hA[kB * kH];    //  8 KB
    __shared__ float hB[kB * kH];    //  8 KB
    __shared__ float g[kB * k4H];    // 32 KB  -> 50 KB total
    int tid = threadIdx.x;
    float cA0 = 0.f, cA1 = 0.f, cB0 = 0.f, cB1 = 0.f;
    for (int i = tid; i < kB * kH; i += 1024) { hA[i] = 0.f; hB[i] = 0.f; }
    __syncthreads();

    for (int t = 0; t < kS; ++t) {
        for (int i = tid; i < kB * kE; i += 1024) {
            int b = i / kE, e = i % kE;
            sX[i] = x[(b * kS + t) * kE + e];
        }
        __syncthreads();
        // layer-0 gates: g[b,j] = x_t . Wih0[j] + hA . Whh0[j] + bih0 + bhh0
        for (int o = tid; o < kB * k4H; o += 1024) {
            int b = o / k4H, j = o % k4H;
            float sum = bih0[j] + bhh0[j];
            const float* wx = Wih0 + j * kE;
            const float* xr = sX + b * kE;
#pragma unroll 8
            for (int e = 0; e < kE; ++e) sum += xr[e] * wx[e];
            const float* wh = Whh0 + j * kH;
            const float* hr = hA + b * kH;
#pragma unroll 8
            for (int e = 0; e < kH; ++e) sum += hr[e] * wh[e];
            g[o] = sum;
        }
        __syncthreads();
        { // layer-0 state update; thread owns (b,u) = tid and tid+1024
            int i = tid;        { int b = i >> 8, u = i & 255; hA[i] = lstm_update(g + b * k4H, u, cA0); }
            i = tid + 1024;     { int b = i >> 8, u = i & 255; hA[i] = lstm_update(g + b * k4H, u, cA1); }
        }
        __syncthreads();
        // layer-1 gates: input is hA (layer-0 output this step)
        for (int o = tid; o < kB * k4H; o += 1024) {
            int b = o / k4H, j = o % k4H;
            float sum = bih1[j] + bhh1[j];
            const float* wi = Wih1 + j * kH;
            const float* ha = hA + b * kH;
#pragma unroll 8
            for (int e = 0; e < kH; ++e) sum += ha[e] * wi[e];
            const float* wh = Whh1 + j * kH;
            const float* hb = hB + b * kH;
#pragma unroll 8
            for (int e = 0; e < kH; ++e) sum += hb[e] * wh[e];
            g[o] = sum;
        }
        __syncthreads();
        { // layer-1 state update + emit sequence
            int i = tid;
            { int b = i >> 8, u = i & 255;
              float h = lstm_update(g + b * k4H, u, cB0);
              hB[i] = h; outseq[(b * kS + t) * kH + u] = h; }
            i = tid + 1024;
            { int b = i >> 8, u = i & 255;
              float h = lstm_update(g + b * k4H, u, cB1);
              hB[i] = h; outseq[(b * kS + t) * kH + u] = h; }
        }
        __syncthreads();
    }
    for (int i = tid; i < kB * kH; i += 1024) h2[i] = hB[i];
}

// ---------------------------------------------------------------------------
// 3. out5 = tanh(outseq @ Wl^T + bl)   [160,256]x[256,320] via WMMA f32 16x16x4
//    200 tiles = 25 blocks x 8 waves.
// ---------------------------------------------------------------------------
__global__ void __launch_bounds__(256)
k_proj5(const float* __restrict__ outseq, const float* __restrict__ Wl,
        const float* __restrict__ bl, float* __restrict__ out5) {
    int wave = threadIdx.x >> 5, lane = threadIdx.x & 31;
    int tile = blockIdx.x * 8 + wave;          // 0..199
    int mT = tile / (kN5 / 16);                // 0..9
    int nT = tile % (kN5 / 16);                // 0..19
    int m0 = mT * 16, n0 = nT * 16;
    int half = lane >> 4, mr = lane & 15;
    v8f acc = {0.f, 0.f, 0.f, 0.f, 0.f, 0.f, 0.f, 0.f};
    const float* Arow = outseq + (m0 + mr) * kH;
    const float* Brow = Wl + (n0 + mr) * kH;   // B[k][n] = Wl[n][k]
    for (int k0 = 0; k0 < kH; k0 += 4) {
        v2f a = *reinterpret_cast<const v2f*>(Arow + k0 + 2 * half);
        v2f b = *reinterpret_cast<const v2f*>(Brow + k0 + 2 * half);
        acc = wmma4(acc, a, b);
    }
#pragma unroll
    for (int vv = 0; vv < 8; ++vv) {
        int m = m0 + vv + 8 * half;
        int n = n0 + mr;
        out5[m * kN5 + n] = tanhf(acc[vv] + bl[n]);
    }
}

// ---------------------------------------------------------------------------
// 4. outputs[b,s,l] = max_k dot(out5[b,s,k,:], video_emb[label[b,s,l],:])
// ---------------------------------------------------------------------------
__global__ void __launch_bounds__(128)
k_outputs(const float* __restrict__ out5, const int* __restrict__ label,
          const float* __restrict__ video_emb, float* __restrict__ outp) {
    int bs = blockIdx.x;                       // 0..159
    __shared__ float s5[5 * kE];
    for (int i = threadIdx.x; i < 5 * kE; i += 128) s5[i] = out5[bs * 5 * kE + i];
    __syncthreads();
    for (int l = threadIdx.x; l < kL; l += 128) {
        int id = label[bs * kL + l];
        const float* em = video_emb + id * kE;
        float a0 = 0.f, a1 = 0.f, a2 = 0.f, a3 = 0.f, a4 = 0.f;
#pragma unroll 8
        for (int e = 0; e < kE; ++e) {
            float ev = em[e];
            a0 += s5[e] * ev;            a1 += s5[kE + e] * ev;
            a2 += s5[2 * kE + e] * ev;   a3 += s5[3 * kE + e] * ev;
            a4 += s5[4 * kE + e] * ev;
        }
        outp[bs * kL + l] = fmaxf(fmaxf(fmaxf(a0, a1), fmaxf(a2, a3)), a4);
    }
}

// ---------------------------------------------------------------------------
// 5. last_out = tanh(h2 @ Wl2^T + bl2)  [8,256]x[256,6400] via WMMA (M padded to 16)
// ---------------------------------------------------------------------------
__global__ void __launch_bounds__(256)
k_lastproj(const float* __restrict__ h2, const float* __restrict__ Wl2,
           const float* __restrict__ bl2, float* __restrict__ last_out) {
    int wave = threadIdx.x >> 5, lane = threadIdx.x & 31;
    int nT = blockIdx.x * 8 + wave;            // 0..399
    int n0 = nT * 16;
    int half = lane >> 4, mr = lane & 15;
    v8f acc = {0.f, 0.f, 0.f, 0.f, 0.f, 0.f, 0.f, 0.f};
    const float* Brow = Wl2 + (n0 + mr) * kH;
    for (int k0 = 0; k0 < kH; k0 += 4) {
        v2f a = {0.f, 0.f};
        if (mr < kB) a = *reinterpret_cast<const v2f*>(h2 + mr * kH + k0 + 2 * half);
        v2f b = *reinterpret_cast<const v2f*>(Brow + k0 + 2 * half);
        acc = wmma4(acc, a, b);
    }
    if (half == 0) {                           // rows 0..7 are the 8 batches
#pragma unroll
        for (int vv = 0; vv < 8; ++vv) {
            int n = n0 + mr;
            last_out[vv * kNU + n] = tanhf(acc[vv] + bl2[n]);
        }
    }
}

// ---------------------------------------------------------------------------
// 6. last_outputs[b,l] = max_u dot(last_out[b,u,:], video_emb[last_label[b,l],:])
// ---------------------------------------------------------------------------
__global__ void __launch_bounds__(128)
k_last_outputs(const float* __restrict__ last_out, const int* __restrict__ last_label,
               const float* __restrict__ video_emb, float* __restrict__ lout) {
    int b = blockIdx.x;
    __shared__ float sA[kU * kE];              // 25.6 KB
    for (int i = threadIdx.x; i < kU * kE; i += 128) sA[i] = last_out[b * kU * kE + i];
    __syncthreads();
    for (int l = threadIdx.x; l < kLL; l += 128) {
        int id = last_label[b * kLL + l];
        const float* em = video_emb + id * kE;
        float r[kE];
#pragma unroll
        for (int e = 0; e < kE; ++e) r[e] = em[e];
        float best = -INFINITY;
        for (int u = 0; u < kU; ++u) {
            float d = 0.f;
#pragma unroll 8
            for (int e = 0; e < kE; ++e) d += sA[u * kE + e] * r[e];
            best = fmaxf(best, d);
        }
        lout[b * kLL + l] = best;
    }
}

// ---------------------------------------------------------------------------
// 7. scores[b,v] = max_u dot(last_out[b,u,:], video_emb[v,:])  -- big WMMA GEMM.
//    One block per 16-column v-tile; tile staged once in LDS (padded stride to
//    avoid bank conflicts) and reused by 8 waves = 8 batches. Padded M rows
//    (100..111) are masked out of the max (init -inf).
// ---------------------------------------------------------------------------
__global__ void __launch_bounds__(256)
k_all_scores(const float* __restrict__ last_out, const float* __restrict__ video_emb,
             float* __restrict__ scores) {
    constexpr int SBS = kE + 4;                // padded row stride (words)
    int v0 = blockIdx.x * 16;
    __shared__ float sBt[16 * SBS];
    {
        int t = threadIdx.x;                   // 256 threads = 256 float4 loads
        int row = t >> 4, c4 = (t & 15) * 4;
        float4 val = *reinterpret_cast<const float4*>(video_emb + (v0 + row) * kE + c4);
        *reinterpret_cast<float4*>(&sBt[row * SBS + c4]) = val;
    }
    if (threadIdx.x < 16 && v0 + 16 < kV) {    // warm L2 for the neighbor tile
        __builtin_prefetch(video_emb + (v0 + 16 + threadIdx.x) * kE, 0, 1);
    }
    __syncthreads();

    int b = threadIdx.x >> 5;                  // wave id == batch id
    int lane = threadIdx.x & 31;
    int half = lane >> 4, mr = lane & 15;
    const float* A = last_out + b * kU * kE;
    float vmax = -INFINITY;
    for (int mT = 0; mT < 7; ++mT) {
        int mrow = mT * 16 + mr;
        bool ok = (mrow < kU);
        const float* Arow = A + mrow * kE;
        v8f acc = {0.f, 0.f, 0.f, 0.f, 0.f, 0.f, 0.f, 0.f};
#pragma unroll 4
        for (int k0 = 0; k0 < kE; k0 += 4) {
            v2f a = {0.f, 0.f};
            if (ok) a = *reinterpret_cast<const v2f*>(Arow + k0 + 2 * half);
            v2f bfr = *reinterpret_cast<const v2f*>(&sBt[mr * SBS + k0 + 2 * half]);
            acc = wmma4(acc, a, bfr);
        }
#pragma unroll
        for (int vv = 0; vv < 8; ++vv) {
            int m = mT * 16 + vv + 8 * half;
            if (m < kU) vmax = fmaxf(vmax, acc[vv]);
        }
    }
    float other = __shfl_xor(vmax, 16, 32);    // lanes L and L+16 share column n
    vmax = fmaxf(vmax, other);
    if (half == 0) scores[b * kV + v0 + mr] = vmax;
}

// ---------------------------------------------------------------------------
// 8. o0[b] = -mean_s( sum_l mask*lt*logsoftmax / (sum_l lt + 1e-10) ) * 0.0
// ---------------------------------------------------------------------------
__global__ void __launch_bounds__(128)
k_logits(const float* __restrict__ outp, const int* __restrict__ mask,
         const float* __restrict__ label_type, float* __restrict__ o0) {
    int b = blockIdx.x, tid = threadIdx.x;
    __shared__ float red[128];
    float acc = 0.f;
    for (int s = 0; s < kS; ++s) {
        int base = (b * kS + s) * kL;
        float x = (tid < kL) ? outp[base + tid] : -INFINITY;
        red[tid] = x; __syncthreads();
        for (int off = 64; off > 0; off >>= 1) { if (tid < off) red[tid] = fmaxf(red[tid], red[tid + off]); __syncthreads(); }
        float mx = red[0]; __syncthreads();
        float ex = (tid < kL) ? expf(x - mx) : 0.f;
        red[tid] = ex; __syncthreads();
        for (int off = 64; off > 0; off >>= 1) { if (tid < off) red[tid] += red[tid + off]; __syncthreads(); }
        float Z = red[0]; __syncthreads();
        float logZ = mx + logf(Z);
        float lt = (tid < kL) ? label_type[base + tid] : 0.f;
        float mf = (float)mask[b * kS + s];
        red[tid] = (tid < kL) ? mf * lt * (x - logZ) : 0.f; __syncthreads();
        for (int off = 64; off > 0; off >>= 1) { if (tid < off) red[tid] += red[tid + off]; __syncthreads(); }
        float num = red[0]; __syncthreads();
        red[tid] = lt; __syncthreads();
        for (int off = 64; off > 0; off >>= 1) { if (tid < off) red[tid] += red[tid + off]; __syncthreads(); }
        float den = red[0]; __syncthreads();
        acc += num / (den + 1e-10f);
    }
    if (tid == 0) o0[b] = -(acc / (float)kS) * 0.0f;
}

// ---------------------------------------------------------------------------
// 9. o1[b] = -0.5 * sum_l lt*logsoftmax(last_outputs) / (sum_l lt + 1e-10)
// ---------------------------------------------------------------------------
__global__ void __launch_bounds__(256)
k_last_logits(const float* __restrict__ lout, const float* __restrict__ llt,
              float* __restrict__ o1) {
    int b = blockIdx.x, tid = threadIdx.x;
    __shared__ float red[256];
    float x = (tid < kLL) ? lout[b * kLL + tid] : -INFINITY;
    red[tid] = x; __syncthreads();
    for (int off = 128; off > 0; off >>= 1) { if (tid < off) red[tid] = fmaxf(red[tid], red[tid + off]); __syncthreads(); }
    float mx = red[0]; __syncthreads();
    float ex = (tid < kLL) ? expf(x - mx) : 0.f;
    red[tid] = ex; __syncthreads();
    for (int off = 128; off > 0; off >>= 1) { if (tid < off) red[tid] += red[tid + off]; __syncthreads(); }
    float Z = red[0]; __syncthreads();
    float logZ = mx + logf(Z);
    float lt = (tid < kLL) ? llt[b * kLL + tid] : 0.f;
    red[tid] = (tid < kLL) ? lt * (x - logZ) : 0.f; __syncthreads();
    for (int off = 128; off > 0; off >>= 1) { if (tid < off) red[tid] += red[tid + off]; __syncthreads(); }
    float num = red[0]; __syncthreads();
    red[tid] = lt; __syncthreads();
    for (int off = 128; off > 0; off >>= 1) { if (tid < off) red[tid] += red[tid + off]; __syncthreads(); }
    float den = red[0];
    if (tid == 0) o1[b] = -0.5f * (num / (den + 1e-10f));
}

// ---------------------------------------------------------------------------
// 10. rec_outputs: K == L == 100 -> full stable argsort (desc value, asc index)
//     via O(L^2) ranking; gather label through the sorted indices.
// ---------------------------------------------------------------------------
__global__ void __launch_bounds__(128)
k_rec(const float* __restrict__ outp, const int* __restrict__ label, float* __restrict__ o2) {
    int bs = blockIdx.x;
    __shared__ float row[kL];
    int tid = threadIdx.x;
    if (tid < kL) row[tid] = outp[bs * kL + tid];
    __syncthreads();
    if (tid < kL) {
        float v = row[tid];
        int r = 0;
        for (int j = 0; j < kL; ++j) {
            float w = row[j];
            r += (w > v) || (w == v && j < tid);
        }
        o2[bs * kL + r] = (float)label[bs * kL + tid];
    }
}

// ---------------------------------------------------------------------------
// 11. last_rec_outputs: top-100 of 200, same ranking trick.
// ---------------------------------------------------------------------------
__global__ void __launch_bounds__(256)
k_last_rec(const float* __restrict__ lout, const int* __restrict__ last_label,
           float* __restrict__ o3) {
    int b = blockIdx.x;
    __shared__ float row[kLL];
    int tid = threadIdx.x;
    if (tid < kLL) row[tid] = lout[b * kLL + tid];
    __syncthreads();
    if (tid < kLL) {
        float v = row[tid];
        int r = 0;
        for (int j = 0; j < kLL; ++j) {
            float w = row[j];
            r += (w > v) || (w == v && j < tid);
        }
        if (r < kK) o3[b * kK + r] = (float)last_label[b * kLL + tid];
    }
}

// ---------------------------------------------------------------------------
// 12. last_rec_all: top-100 of 100000 per batch via 4-pass radix select on
//     monotone float keys, exact JAX tie-break (lower index first).
// ---------------------------------------------------------------------------
__global__ void __launch_bounds__(1024)
k_topk_all(const float* __restrict__ scores, float* __restrict__ o4) {
    int b = blockIdx.x;
    const float* sc = scores + b * kV;
    __shared__ unsigned hist[256];
    __shared__ unsigned sPref;
    __shared__ int sNeed;
    __shared__ unsigned cntG, cntE;
    __shared__ unsigned selKey[128];
    __shared__ int selIdx[128];
    __shared__ int eqIdx[1024];
    int tid = threadIdx.x;
    if (tid == 0) { sPref = 0u; sNeed = kK; }
    __syncthreads();
    for (int shift = 24; shift >= 0; shift -= 8) {
        for (int i = tid; i < 256; i += 1024) hist[i] = 0u;
        __syncthreads();
        unsigned pref = sPref;
        unsigned pm = (shift == 24) ? 0u : (0xFFFFFFFFu << (shift + 8));
        for (int v = tid; v < kV; v += 1024) {
            unsigned k = fkey(sc[v]);
            if ((k & pm) == (pref & pm)) atomicAdd(&hist[(k >> shift) & 255u], 1u);
        }
        __syncthreads();
        if (tid == 0) {
            int need = sNeed;
            unsigned cum = 0;
            int bin = 0;
            for (bin = 255; bin >= 0; --bin) {
                unsigned h = hist[bin];
                if (cum + h >= (unsigned)need) break;
                cum += h;
            }
            if (bin < 0) bin = 0;
            sPref = pref | ((unsigned)bin << shift);
            sNeed = need - (int)cum;
        }
        __syncthreads();
    }
    unsigned T = sPref;
    int need = sNeed;                         // #elements == T to include
    if (tid == 0) { cntG = 0u; cntE = 0u; }
    __syncthreads();
    for (int v = tid; v < kV; v += 1024) {
        unsigned k = fkey(sc[v]);
        if (k > T) {
            unsigned p = atomicAdd(&cntG, 1u);
            if (p < 128u) { selKey[p] = k; selIdx[p] = v; }
        } else if (k == T) {
            unsigned p = atomicAdd(&cntE, 1u);
            if (p < 1024u) eqIdx[p] = v;
        }
    }
    __syncthreads();
    int nG = (cntG < 128u) ? (int)cntG : 128;
    int nE = (cntE < 1024u) ? (int)cntE : 1024;
    for (int i = tid; i < nE; i += 1024) {    // smallest `need` equal indices
        int idx = eqIdx[i];
        int r = 0;
        for (int j = 0; j < nE; ++j) r += (eqIdx[j] < idx);
        if (r < need && (nG + r) < 128) { selKey[nG + r] = T; selIdx[nG + r] = idx; }
    }
    __syncthreads();
    int total = nG + need;
    if (total > kK) total = kK;
    for (int i = tid; i < total; i += 1024) { // final order: desc key, asc index
        unsigned ki = selKey[i];
        int ii = selIdx[i];
        int r = 0;
        for (int j = 0; j < total; ++j) {
            unsigned kj = selKey[j];
            r += (kj > ki) || (kj == ki && selIdx[j] < ii);
        }
        if (r < kK) o4[b * kK + r] = (float)ii;
    }
}

// ---------------------------------------------------------------------------
// Launcher
// ---------------------------------------------------------------------------
extern "C" void kernel_launch(void* const* d_in, const int* in_sizes, int n_in,
                              void* d_out, int out_size, void* d_ws, size_t ws_size,
                              hipStream_t stream) {
    (void)in_sizes; (void)n_in; (void)out_size; (void)ws_size;

    const int*   inputs     = (const int*)d_in[0];
    const int*   mask       = (const int*)d_in[1];
    const int*   label      = (const int*)d_in[2];
    const float* label_type = (const float*)d_in[3];
    const int*   last_label = (const int*)d_in[4];
    const float* last_label_type = (const float*)d_in[5];
    const float* graph_emb  = (const float*)d_in[6];
    const float* video_emb  = (const float*)d_in[7];
    const float* Wih0 = (const float*)d_in[8];
    const float* Whh0 = (const float*)d_in[9];
    const float* bih0 = (const float*)d_in[10];
    const float* bhh0 = (const float*)d_in[11];
    const float* Wih1 = (const float*)d_in[12];
    const float* Whh1 = (const float*)d_in[13];
    const float* bih1 = (const float*)d_in[14];
    const float* bhh1 = (const float*)d_in[15];
    const float* Wl   = (const float*)d_in[16];
    const float* bl   = (const float*)d_in[17];
    const float* Wl2  = (const float*)d_in[18];
    const float* bl2  = (const float*)d_in[19];

    // workspace layout (floats), ~3.9 MB total, fully rewritten every call
    float* ws      = (float*)d_ws;
    float* x       = ws;                      // B*S*E      = 10240
    float* outseq  = x + kB * kS * kE;        // B*S*H      = 40960
    float* h2      = outseq + kB * kS * kH;   // B*H        =  2048
    float* out5    = h2 + kB * kH;            // B*S*5*E    = 51200
    float* outp    = out5 + kB * kS * kN5;    // B*S*L      = 16000
    float* lastout = outp + kB * kS * kL;     // B*U*E      = 51200
    float* lout    = lastout + kB * kNU;      // B*LL       =  1600
    float* scores  = lout + kB * kLL;         // B*V        = 800000

    float* o0 = (float*)d_out;                // [B]
    float* o1 = o0 + kB;                      // [B]
    float* o2 = o1 + kB;                      // [B,S,K]
    float* o3 = o2 + kB * kS * kK;            // [B,K]
    float* o4 = o3 + kB * kK;                 // [B,K]

    k_gather<<<(kB * kS * kE + 255) / 256, 256, 0, stream>>>(inputs, mask, graph_emb, x);
    k_lstm<<<1, 1024, 0, stream>>>(x, Wih0, Whh0, bih0, bhh0, Wih1, Whh1, bih1, bhh1,
                                   outseq, h2);
    k_proj5<<<25, 256, 0, stream>>>(outseq, Wl, bl, out5);
    k_outputs<<<kB * kS, 128, 0, stream>>>(out5, label, video_emb, outp);
    k_lastproj<<<50, 256, 0, stream>>>(h2, Wl2, bl2, lastout);
    k_last_outputs<<<kB, 128, 0, stream>>>(lastout, last_label, video_emb, lout);
    k_all_scores<<<kV / 16, 256, 0, stream>>>(lastout, video_emb, scores);
    k_logits<<<kB, 128, 0, stream>>>(outp, mask, label_type, o0);
    k_last_logits<<<kB, 256, 0, stream>>>(lout, last_label_type, o1);
    k_rec<<<kB * kS, 128, 0, stream>>>(outp, label, o2);
    k_last_rec<<<kB, 256, 0, stream>>>(lout, last_label, o3);
    k_topk_all<<<kB, 1024, 0, stream>>>(scores, o4);
}